// SAGPool_mix_4544075399264
// MI455X (gfx1250) — compile-verified
//
#include <hip/hip_runtime.h>
#include <hip/hip_bf16.h>

#define N_NODES 100000
#define N_EDGES 3200000
#define DFEAT   512
#define KKEEP   50000
#define NTILES  (N_NODES / 16)     // 6250, exact
#define NBUCKET 65536

typedef __attribute__((ext_vector_type(2))) float v2f;
typedef __attribute__((ext_vector_type(8))) float v8f;

// float -> uint key such that ascending uint order == descending float order,
// ties broken later by ascending index (matches jax.lax.top_k).
__device__ __forceinline__ unsigned sortKeyDesc(float f) {
  unsigned u = __float_as_uint(f);
  u ^= (u & 0x80000000u) ? 0xFFFFFFFFu : 0x80000000u; // ascending order key
  return ~u;                                          // flip -> descending
}

// ---------------- h = x @ W via V_WMMA_F32_16X16X4_F32 -------------------
// One wave per 16-node tile. A = x[tile,:] 16x4 slab; B = W[k:k+4] broadcast
// into all 16 columns. A-layout (ISA 7.12.2, 32-bit A 16x4): lane<16 holds
// K=0,1; lane>=16 holds K=2,3 for M=lane&15. B mirrors (K=0,1 | K=2,3 per
// lane half). C layout: VGPR r = rows r (lanes 0-15) and r+8 (lanes 16-31);
// with broadcast-B every lane in a half holds identical row sums.
__global__ void k_gemv_wmma(const float* __restrict__ x,
                            const float* __restrict__ W,
                            float* __restrict__ h) {
  int lane = threadIdx.x & 31;
  int wave = (int)((blockIdx.x * blockDim.x + threadIdx.x) >> 5);
  if (wave >= NTILES) return;                 // uniform per-wave exit
  int row   = wave * 16 + (lane & 15);
  int kband = (lane >> 4) << 1;               // 0 or 2
  const float* xp = x + (size_t)row * DFEAT + kband;
  const float* wp = W + kband;
  v8f c = {};
#pragma unroll 4
  for (int k0 = 0; k0 < DFEAT; k0 += 4) {
    v2f a = *(const v2f*)(xp + k0);
    v2f b = *(const v2f*)(wp + k0);
    c = __builtin_amdgcn_wmma_f32_16x16x4_f32(false, a, false, b,
                                              (short)0, c, false, false);
  }
  if ((lane & 15) == 0) {                     // lane 0 -> rows 0..7, lane 16 -> 8..15
    int base = wave * 16 + (lane >> 4) * 8;
#pragma unroll
    for (int r = 0; r < 8; ++r) h[base + r] = c[r];
  }
}

// ---------------- init: deg=1 (self loop), hist=0, mask=0 ----------------
__global__ void k_init(unsigned* __restrict__ udeg, unsigned* __restrict__ hist,
                       float* __restrict__ mask) {
  int i = blockIdx.x * blockDim.x + threadIdx.x;
  if (i < N_NODES) { udeg[i] = 1u; mask[i] = 0.0f; }
  if (i < NBUCKET) hist[i] = 0u;
}

__global__ void k_deg(const int* __restrict__ ei, unsigned* __restrict__ udeg) {
  int e = blockIdx.x * blockDim.x + threadIdx.x;
  if (e < N_EDGES) atomicAdd(&udeg[ei[N_EDGES + e]], 1u);   // col
}

__global__ void k_dinv_self(const unsigned* __restrict__ udeg,
                            const float* __restrict__ h,
                            float* __restrict__ dinv, float* __restrict__ s) {
  int v = blockIdx.x * blockDim.x + threadIdx.x;
  if (v < N_NODES) {
    float di = rsqrtf((float)udeg[v]);        // deg >= 1 always
    dinv[v] = di;
    s[v] = di * di * h[v];                    // self-loop contribution
  }
}

__global__ void k_edge_score(const int* __restrict__ ei,
                             const float* __restrict__ dinv,
                             const float* __restrict__ h,
                             float* __restrict__ s) {
  int e = blockIdx.x * blockDim.x + threadIdx.x;
  if (e < N_EDGES) {
    int r = ei[e], c = ei[N_EDGES + e];
    atomicAdd(&s[c], dinv[r] * dinv[c] * h[r]);
  }
}

__global__ void k_score(const float* __restrict__ s, const float* __restrict__ b,
                        float* __restrict__ score, unsigned* __restrict__ key,
                        unsigned* __restrict__ hist) {
  int v = blockIdx.x * blockDim.x + threadIdx.x;
  if (v < N_NODES) {
    float sc = tanhf(s[v] + b[0]);
    score[v] = sc;
    unsigned k = sortKeyDesc(sc);
    key[v] = k;
    atomicAdd(&hist[k >> 16], 1u);
  }
}

// exclusive scan of 65536 bins in one 1024-thread block (64 bins/thread)
__global__ void k_scan(const unsigned* __restrict__ hist,
                       unsigned* __restrict__ bstart,
                       unsigned* __restrict__ cursor) {
  __shared__ unsigned sm[1024];
  int t = threadIdx.x, base = t * 64;
  unsigned sum = 0;
  for (int i = 0; i < 64; ++i) sum += hist[base + i];
  sm[t] = sum;
  __syncthreads();
  for (int off = 1; off < 1024; off <<= 1) {
    unsigned v = (t >= off) ? sm[t - off] : 0u;
    __syncthreads();
    sm[t] += v;
    __syncthreads();
  }
  unsigned run = (t == 0) ? 0u : sm[t - 1];
  for (int i = 0; i < 64; ++i) {
    unsigned hv = hist[base + i];
    bstart[base + i] = run;
    cursor[base + i] = run;
    run += hv;
  }
  if (t == 1023) bstart[NBUCKET] = run;       // == N_NODES
}

__global__ void k_scatter(const unsigned* __restrict__ key,
                          unsigned* __restrict__ cursor,
                          unsigned* __restrict__ bkey,
                          unsigned* __restrict__ bidx) {
  int v = blockIdx.x * blockDim.x + threadIdx.x;
  if (v < N_NODES) {
    unsigned k = key[v];
    unsigned p = atomicAdd(&cursor[k >> 16], 1u);
    bkey[p] = k;
    bidx[p] = (unsigned)v;
  }
}

// rank within bucket by counting strictly-smaller (key, idx); order-independent,
// so the nondeterministic scatter order inside a bucket cannot change results.
__global__ void k_rank(const unsigned* __restrict__ bkey,
                       const unsigned* __restrict__ bidx,
                       const unsigned* __restrict__ bstart,
                       float* __restrict__ mask, float* __restrict__ out_perm) {
  int p = blockIdx.x * blockDim.x + threadIdx.x;
  if (p >= N_NODES) return;
  unsigned k = bkey[p], i = bidx[p];
  unsigned b = k >> 16;
  unsigned st = bstart[b], en = bstart[b + 1];
  unsigned cnt = 0;
  for (unsigned q = st; q < en; ++q) {
    unsigned kk = bkey[q];
    cnt += (kk < k) || (kk == k && bidx[q] < i);
  }
  unsigned pos = st + cnt;                    // final sorted position
  if (pos < KKEEP) {
    mask[i] = 1.0f;
    out_perm[pos] = (float)i;
  }
}

__global__ void k_scale_x(const float* __restrict__ x,
                          const float* __restrict__ score,
                          const float* __restrict__ mask,
                          float* __restrict__ out) {
  int idx = blockIdx.x * blockDim.x + threadIdx.x;  // float4 index, < 12.8M
  if (idx < N_NODES * (DFEAT / 4)) {
    int row = idx / (DFEAT / 4);
    float sc = score[row] * mask[row];
    float4 xv = ((const float4*)x)[idx];
    float4 o  = make_float4(xv.x * sc, xv.y * sc, xv.z * sc, xv.w * sc);
    ((float4*)out)[idx] = o;
  }
}

__global__ void k_edge_out(const int* __restrict__ ei,
                           const float* __restrict__ mask,
                           float* __restrict__ out_ei,
                           float* __restrict__ out_ew) {
  int e = blockIdx.x * blockDim.x + threadIdx.x;
  if (e < N_EDGES) {
    int r = ei[e], c = ei[N_EDGES + e];
    out_ei[e] = (float)r;
    out_ei[N_EDGES + e] = (float)c;
    out_ew[e] = mask[r] * mask[c];            // 1 iff both kept
  }
}

__global__ void k_batch(float* __restrict__ out_batch) {
  int v = blockIdx.x * blockDim.x + threadIdx.x;
  if (v < N_NODES) out_batch[v] = 0.0f;
}

extern "C" void kernel_launch(void* const* d_in, const int* in_sizes, int n_in,
                              void* d_out, int out_size, void* d_ws, size_t ws_size,
                              hipStream_t stream) {
  const float* x  = (const float*)d_in[0];
  const int*   ei = (const int*)d_in[1];
  const float* W  = (const float*)d_in[2];
  const float* b  = (const float*)d_in[3];

  // workspace layout (4-byte units)
  char* wsb = (char*)d_ws;
  size_t off = 0;
  auto grab = [&](size_t elems) { void* p = wsb + off; off += elems * 4; return p; };
  float*    h      = (float*)grab(N_NODES);
  unsigned* udeg   = (unsigned*)grab(N_NODES);
  float*    dinv   = (float*)grab(N_NODES);
  float*    s      = (float*)grab(N_NODES);
  float*    score  = (float*)grab(N_NODES);
  unsigned* key    = (unsigned*)grab(N_NODES);
  unsigned* hist   = (unsigned*)grab(NBUCKET);
  unsigned* bstart = (unsigned*)grab(NBUCKET + 1);
  unsigned* cursor = (unsigned*)grab(NBUCKET);
  unsigned* bkey   = (unsigned*)grab(N_NODES);
  unsigned* bidx   = (unsigned*)grab(N_NODES);
  float*    mask   = (float*)grab(N_NODES);
  (void)ws_size;

  // output layout (flat float, tuple order)
  float* out       = (float*)d_out;
  float* out_x2    = out;
  float* out_ei    = out_x2 + (size_t)N_NODES * DFEAT;
  float* out_ew    = out_ei + 2 * (size_t)N_EDGES;
  float* out_batch = out_ew + (size_t)N_EDGES;
  float* out_perm  = out_batch + (size_t)N_NODES;
  (void)out_size; (void)n_in; (void)in_sizes;

  const int T = 256;
  const int gN = (N_NODES + T - 1) / T;          // 391
  const int gE = (N_EDGES + T - 1) / T;          // 12500
  const int gW = (NTILES * 32 + T - 1) / T;      // 782 (one wave per 16-row tile)
  const int gX = (N_NODES * (DFEAT / 4) + T - 1) / T;  // 50000

  k_init<<<gN, T, 0, stream>>>(udeg, hist, mask);
  k_gemv_wmma<<<gW, T, 0, stream>>>(x, W, h);
  k_deg<<<gE, T, 0, stream>>>(ei, udeg);
  k_dinv_self<<<gN, T, 0, stream>>>(udeg, h, dinv, s);
  k_edge_score<<<gE, T, 0, stream>>>(ei, dinv, h, s);
  k_score<<<gN, T, 0, stream>>>(s, b, score, key, hist);
  k_scan<<<1, 1024, 0, stream>>>(hist, bstart, cursor);
  k_scatter<<<gN, T, 0, stream>>>(key, cursor, bkey, bidx);
  k_rank<<<gN, T, 0, stream>>>(bkey, bidx, bstart, mask, out_perm);
  k_scale_x<<<gX, T, 0, stream>>>(x, score, mask, out_x2);
  k_edge_out<<<gE, T, 0, stream>>>(ei, mask, out_ei, out_ew);
  k_batch<<<gN, T, 0, stream>>>(out_batch);
}